// LSTM_5471788335631
// MI455X (gfx1250) — compile-verified
//
#include <hip/hip_runtime.h>
#include <hip/hip_bf16.h>

// LSTM: S=512, B=64, I=1024, H=1024  (all fp32)
//   xW = einsum('sbi,gi->sbg', x, W_w) + W_b         (parallel GEMM, WMMA f32)
//   recurrent scan over S with gates = xW[t] + h @ U_w^T   (persistent kernel)

#define S_LEN   512
#define BATCH   64
#define IN_DIM  1024
#define HID     1024
#define GATES   4096          // 4*HID
#define LDS_STRIDE 1028       // padded row stride (floats) -> bank-conflict-free, 16B aligned

typedef float v2f __attribute__((ext_vector_type(2)));
typedef float v8f __attribute__((ext_vector_type(8)));
typedef int   v4i __attribute__((ext_vector_type(4)));

#if __has_builtin(__builtin_amdgcn_global_load_async_to_lds_b128) && \
    __has_builtin(__builtin_amdgcn_s_wait_asynccnt)
#define HAVE_ASYNC_LDS 1
#else
#define HAVE_ASYNC_LDS 0
#endif

typedef __attribute__((address_space(1))) v4i* gptr_v4i;
typedef __attribute__((address_space(3))) v4i* lptr_v4i;

__device__ __forceinline__ v8f wmma_f32(v2f a, v2f b, v8f c) {
  // V_WMMA_F32_16X16X4_F32 : D = A(16x4) * B(4x16) + C(16x16), full fp32
  return __builtin_amdgcn_wmma_f32_16x16x4_f32(false, a, false, b, (short)0, c,
                                               false, false);
}

__device__ __forceinline__ float fsigmoid(float x) {
  return 1.0f / (1.0f + __expf(-x));
}

// -------------------------------------------------------------------------
// Kernel 1: xw[(s*B+b), g] = sum_k x[s,b,k] * W_w[g,k] + W_b[g]
// Wave computes a 16(M) x 64(N) strip: one A fragment reused across 4 B tiles.
// A lane layout (16x4 f32): lanes 0-15 hold M=lane,K={k,k+1}; lanes 16-31 K={k+2,k+3}
// B lane layout (4x16 f32): lanes 0-15 hold N=lane,K={k,k+1}; lanes 16-31 K={k+2,k+3}
// -------------------------------------------------------------------------
__global__ __launch_bounds__(128)
void lstm_xw_gemm(const float* __restrict__ x, const float* __restrict__ Ww,
                  const float* __restrict__ Wb, float* __restrict__ xw) {
  const int lane  = threadIdx.x & 31;
  const int wave  = threadIdx.x >> 5;     // 0..3
  const int l15   = lane & 15;
  const int hi    = lane >> 4;            // 0 or 1
  const int koff  = hi * 2;
  const int rowoff = hi * 8;
  const int Mbase = blockIdx.y * 64 + wave * 16;
  const int Nbase = blockIdx.x * 64;

  v8f acc[4];
#pragma unroll
  for (int nt = 0; nt < 4; ++nt)
#pragma unroll
    for (int r = 0; r < 8; ++r) acc[nt][r] = 0.0f;

  const float* arow = x + (size_t)(Mbase + l15) * IN_DIM + koff;
  const float* brow0 = Ww + (size_t)(Nbase +  0 + l15) * IN_DIM + koff;
  const float* brow1 = Ww + (size_t)(Nbase + 16 + l15) * IN_DIM + koff;
  const float* brow2 = Ww + (size_t)(Nbase + 32 + l15) * IN_DIM + koff;
  const float* brow3 = Ww + (size_t)(Nbase + 48 + l15) * IN_DIM + koff;

#pragma unroll 4
  for (int k0 = 0; k0 < IN_DIM; k0 += 4) {
    v2f a  = *(const v2f*)(arow  + k0);
    v2f b0 = *(const v2f*)(brow0 + k0);
    v2f b1 = *(const v2f*)(brow1 + k0);
    v2f b2 = *(const v2f*)(brow2 + k0);
    v2f b3 = *(const v2f*)(brow3 + k0);
    acc[0] = wmma_f32(a, b0, acc[0]);
    acc[1] = wmma_f32(a, b1, acc[1]);
    acc[2] = wmma_f32(a, b2, acc[2]);
    acc[3] = wmma_f32(a, b3, acc[3]);
  }

  // C/D layout: VGPR r -> (row = r + 8*hi, col = l15)
#pragma unroll
  for (int nt = 0; nt < 4; ++nt) {
    const int col  = Nbase + nt * 16 + l15;
    const float bias = Wb[col];
#pragma unroll
    for (int r = 0; r < 8; ++r) {
      const int row = Mbase + r + rowoff;
      xw[(size_t)row * GATES + col] = acc[nt][r] + bias;
    }
  }
}

// -------------------------------------------------------------------------
// Kernel 2: persistent recurrent scan. 64 WGs x 256 threads (8 waves).
// WG g owns hidden columns [g*16, g*16+16): keeps the 64 matching U rows in
// LDS (~257 KB, filled via async global->LDS copies) for all 512 steps.
// Grid-wide barrier per step via a monotonic atomic counter. c state stays
// in registers. Next step's xW seed tile is software-prefetched.
// -------------------------------------------------------------------------
extern __shared__ float sLds[];

__global__ __launch_bounds__(256)
void lstm_recurrent(const float* __restrict__ xw, const float* __restrict__ h0,
                    const float* __restrict__ c0, const float* __restrict__ Uw,
                    float* __restrict__ out, float* __restrict__ hbuf,
                    unsigned* __restrict__ syncCnt) {
  const int g    = blockIdx.x;          // 0..63 hidden-slice owner
  const int tid  = threadIdx.x;
  const int lane = tid & 31;
  const int wave = tid >> 5;            // 0..7
  const int l15  = lane & 15;
  const int hi   = lane >> 4;
  const int koff = hi * 2;
  const int rowoff = hi * 8;

  float* Ulds    = sLds;                        // 64 rows x LDS_STRIDE floats
  float* scratch = sLds + 64 * LDS_STRIDE;      // 4 mtiles x 2 tiles x 256 floats

  // --- stage U rows (gate*H + g*16 + j, :) into LDS, 16B chunks ---
  for (int idx = tid; idx < 64 * (HID / 4); idx += 256) {
    const int lr   = idx >> 8;                  // local row 0..63
    const int k4   = idx & 255;
    const int gate = lr >> 4;
    const int j    = lr & 15;
    const float* src = Uw + ((size_t)(gate * HID + g * 16 + j)) * HID + (size_t)k4 * 4;
    float*       dst = Ulds + (size_t)lr * LDS_STRIDE + k4 * 4;
#if HAVE_ASYNC_LDS
    // ASYNCcnt-tracked DMA: GLOBAL_LOAD_ASYNC_TO_LDS_B128 (no VGPR round trip)
    __builtin_amdgcn_global_load_async_to_lds_b128((gptr_v4i)src, (lptr_v4i)dst,
                                                   0, 0);
#else
    *(float4*)dst = *(const float4*)src;
#endif
  }
  // --- init h ping buffer slice (our 16 columns) ---
  for (int idx = tid; idx < BATCH * 16; idx += 256) {
    const int b = idx >> 4, j = idx & 15;
    hbuf[b * HID + g * 16 + j] = h0[b * HID + g * 16 + j];
  }
#if HAVE_ASYNC_LDS
  __builtin_amdgcn_s_wait_asynccnt(0);
#endif

  const int mt    = wave >> 1;            // batch tile 0..3
  const int Mb    = mt * 16;
  const int gbase = (wave & 1) * 2;       // even waves: gates {0,1}=i,f ; odd: {2,3}=g,o
  const int colH  = g * 16 + l15;         // hidden column this lane owns

  // cell state in registers (meaningful on even waves)
  v8f creg;
#pragma unroll
  for (int r = 0; r < 8; ++r)
    creg[r] = c0[(size_t)(Mb + r + rowoff) * HID + colH];

  unsigned barGen = 0;
  auto gsync = [&]() {
    __threadfence();
    __syncthreads();
    if (tid == 0) {
      barGen += gridDim.x;
      atomicAdd(syncCnt, 1u);
      while (__hip_atomic_load(syncCnt, __ATOMIC_RELAXED,
                               __HIP_MEMORY_SCOPE_AGENT) < barGen)
        __builtin_amdgcn_s_sleep(1);
    }
    __syncthreads();
    __threadfence();
  };

  gsync();  // all U staged, all h0 slices visible

  const float* hcur  = hbuf;
  float*       hnext = hbuf + BATCH * HID;

  for (int t = 0; t < S_LEN; ++t) {
    // accumulators seeded from precomputed xW (bias already folded in)
    v8f acc0, acc1;
    const size_t xwbase = (size_t)(t * BATCH) * GATES;
#pragma unroll
    for (int r = 0; r < 8; ++r) {
      const size_t row = (size_t)(Mb + r + rowoff);
      acc0[r] = xw[xwbase + row * GATES + ((gbase + 0) * HID + colH)];
      acc1[r] = xw[xwbase + row * GATES + ((gbase + 1) * HID + colH)];
    }

    // software prefetch of next step's seed tile (hides HBM latency behind
    // the 256-iteration WMMA K-loop; emits global_prefetch)
    if (t + 1 < S_LEN) {
      const float* nxt = xw + (size_t)((t + 1) * BATCH) * GATES;
#pragma unroll
      for (int r = 0; r < 8; ++r) {
        const size_t row = (size_t)(Mb + r + rowoff);
        __builtin_prefetch(nxt + row * GATES + ((gbase + 0) * HID + colH), 0, 1);
        __builtin_prefetch(nxt + row * GATES + ((gbase + 1) * HID + colH), 0, 1);
      }
    }

    const float* arow  = hcur + (size_t)(Mb + l15) * HID + koff;
    const float* b0row = Ulds + (size_t)((gbase + 0) * 16 + l15) * LDS_STRIDE + koff;
    const float* b1row = Ulds + (size_t)((gbase + 1) * 16 + l15) * LDS_STRIDE + koff;

#pragma unroll 4
    for (int k0 = 0; k0 < HID; k0 += 4) {
      v2f a  = *(const v2f*)(arow  + k0);   // h fragment (global, L2-resident)
      v2f b0 = *(const v2f*)(b0row + k0);   // U fragments (LDS, conflict-free)
      v2f b1 = *(const v2f*)(b1row + k0);
      acc0 = wmma_f32(a, b0, acc0);
      acc1 = wmma_f32(a, b1, acc1);
    }

    if (wave & 1) {
      // odd waves hand g/o gate tiles to partner via LDS scratch
      float* dst = scratch + mt * 512;
#pragma unroll
      for (int r = 0; r < 8; ++r) {
        dst[(r + rowoff) * 16 + l15]       = acc0[r];   // gate 2 (g)
        dst[256 + (r + rowoff) * 16 + l15] = acc1[r];   // gate 3 (o)
      }
    }
    __syncthreads();

    if (!(wave & 1)) {
      const float* src = scratch + mt * 512;
#pragma unroll
      for (int r = 0; r < 8; ++r) {
        const float ig = fsigmoid(acc0[r]);
        const float fg = fsigmoid(acc1[r]);
        const float gg = tanhf(src[(r + rowoff) * 16 + l15]);
        const float og = fsigmoid(src[256 + (r + rowoff) * 16 + l15]);
        const float c  = fg * creg[r] + ig * gg;
        const float h  = og * tanhf(c);
        creg[r] = c;
        const int row = Mb + r + rowoff;
        hnext[row * HID + colH] = h;
        out[(size_t)t * (BATCH * HID) + (size_t)row * HID + colH] = h;
        if (t == S_LEN - 1) {
          out[(size_t)S_LEN * BATCH * HID + (size_t)row * HID + colH] = h;  // h_t
          out[(size_t)S_LEN * BATCH * HID + BATCH * HID +
              (size_t)row * HID + colH] = c;                               // c_t
        }
      }
    }

    // swap ping-pong h buffers, then device-wide step barrier
    const float* tmp = hcur; hcur = hnext; hnext = (float*)tmp;
    gsync();
  }
}

// -------------------------------------------------------------------------
extern "C" void kernel_launch(void* const* d_in, const int* in_sizes, int n_in,
                              void* d_out, int out_size, void* d_ws, size_t ws_size,
                              hipStream_t stream) {
  const float* x  = (const float*)d_in[0];  // [S,B,I]
  const float* h0 = (const float*)d_in[1];  // [B,H]
  const float* c0 = (const float*)d_in[2];  // [B,H]
  const float* Ww = (const float*)d_in[3];  // [4H,I]
  const float* Wb = (const float*)d_in[4];  // [4H]
  const float* Uw = (const float*)d_in[5];  // [4H,H]
  float* out = (float*)d_out;               // [S,B,H] ++ [B,H] ++ [B,H]

  // workspace: xW (512 MB) | h ping-pong (2x256 KB) | sync counter
  float*    xw      = (float*)d_ws;
  float*    hbuf    = xw + (size_t)S_LEN * BATCH * GATES;
  unsigned* syncCnt = (unsigned*)(hbuf + 2 * BATCH * HID);

  (void)hipMemsetAsync(syncCnt, 0, sizeof(unsigned), stream);  // capture-safe

  dim3 g1(GATES / 64, (S_LEN * BATCH) / 64);
  lstm_xw_gemm<<<g1, 128, 0, stream>>>(x, Ww, Wb, xw);

  const size_t ldsBytes = (size_t)(64 * LDS_STRIDE + 4 * 512) * sizeof(float);
  lstm_recurrent<<<64, 256, ldsBytes, stream>>>(xw, h0, c0, Uw, out, hbuf, syncCnt);
}